// FlashLM_v73_10428180594727
// MI455X (gfx1250) — compile-verified
//
#include <hip/hip_runtime.h>
#include <hip/hip_bf16.h>
#include <math.h>

typedef __attribute__((ext_vector_type(2))) float v2f;
typedef __attribute__((ext_vector_type(8))) float v8f;

#define TT 2048
#define DD 512
#define DFF 2048
#define NH 8
#define DH 64
#define WIN 256
#define KC 15
#define DM 64
#define VOCAB 32000

__device__ __forceinline__ v8f wmma4(v2f a, v2f b, v8f c) {
  // D = A(16x4 f32) * B(4x16 f32) + C(16x16 f32)
  return __builtin_amdgcn_wmma_f32_16x16x4_f32(false, a, false, b, (short)0, c, false, false);
}

__device__ __forceinline__ float sigf(float x) { return 1.0f / (1.0f + __expf(-x)); }

// ------------------------- reductions -------------------------
__device__ __forceinline__ float block_sum_256(float v) {
  #pragma unroll
  for (int m = 16; m >= 1; m >>= 1) v += __shfl_xor(v, m, 32);
  __shared__ float red[8];
  if ((threadIdx.x & 31) == 0) red[threadIdx.x >> 5] = v;
  __syncthreads();
  float t = red[0] + red[1] + red[2] + red[3] + red[4] + red[5] + red[6] + red[7];
  __syncthreads();
  return t;
}

// ------------------------- RMS norm -------------------------
__global__ __launch_bounds__(256) void rms_kernel(const float* __restrict__ x,
                                                  const float* __restrict__ w,
                                                  float* __restrict__ o) {
  const int t = blockIdx.x, i = threadIdx.x;
  const float* row = x + (size_t)t * DD;
  float v0 = row[i], v1 = row[i + 256];
  float ss = block_sum_256(v0 * v0 + v1 * v1);
  float inv = rsqrtf(ss * (1.0f / DD) + 1e-6f);
  o[(size_t)t * DD + i]       = v0 * inv * w[i];
  o[(size_t)t * DD + i + 256] = v1 * inv * w[i + 256];
}

__global__ __launch_bounds__(256) void embed_rms_kernel(const int* __restrict__ ids,
                                                        const float* __restrict__ emb,
                                                        const float* __restrict__ w,
                                                        float* __restrict__ o) {
  const int t = blockIdx.x, i = threadIdx.x;
  const float* row = emb + (size_t)ids[t] * DD;
  float v0 = row[i], v1 = row[i + 256];
  float ss = block_sum_256(v0 * v0 + v1 * v1);
  float inv = rsqrtf(ss * (1.0f / DD) + 1e-6f);
  o[(size_t)t * DD + i]       = v0 * inv * w[i];
  o[(size_t)t * DD + i + 256] = v1 * inv * w[i + 256];
}

// ------------------------- WMMA GEMM: C[M,N] = A[M,K] * W[N,K]^T (+R) -------------------------
// grid = (M/256, N/64), block = 256 (8 waves); wave w -> rows [bx*256+w*32, +32), cols [by*64, +64)
// Each wave: 32x64 tile = 2 A-frags sharing 4 B-frags per k-step (8 WMMA : 6 b64 loads).
__global__ __launch_bounds__(256) void gemm_kernel(const float* __restrict__ A, int lda,
                                                   const float* __restrict__ W, int ldw,
                                                   float* C, int ldc,
                                                   const float* R, int K) {
  const int lane = threadIdx.x & 31;
  const int wave = threadIdx.x >> 5;
  const int nlo  = lane & 15;
  const int hi   = lane >> 4;
  const int koff = hi << 1;
  const int m0 = blockIdx.x * 256 + wave * 32;
  const int n0 = blockIdx.y * 64;

  const float* a0p = A + (size_t)(m0 + nlo) * lda + koff;
  const float* a1p = a0p + (size_t)16 * lda;
  const float* w0 = W + (size_t)(n0 + nlo) * ldw + koff;
  const float* w1 = w0 + (size_t)16 * ldw;
  const float* w2 = w0 + (size_t)32 * ldw;
  const float* w3 = w0 + (size_t)48 * ldw;

  v8f z = {0,0,0,0,0,0,0,0};
  v8f acc00 = z, acc01 = z, acc02 = z, acc03 = z;
  v8f acc10 = z, acc11 = z, acc12 = z, acc13 = z;

  #pragma unroll 4
  for (int k = 0; k < K; k += 4) {
    v2f a0 = *(const v2f*)(a0p + k);
    v2f a1 = *(const v2f*)(a1p + k);
    v2f b0 = *(const v2f*)(w0 + k);
    v2f b1 = *(const v2f*)(w1 + k);
    v2f b2 = *(const v2f*)(w2 + k);
    v2f b3 = *(const v2f*)(w3 + k);
    acc00 = wmma4(a0, b0, acc00);
    acc10 = wmma4(a1, b0, acc10);
    acc01 = wmma4(a0, b1, acc01);
    acc11 = wmma4(a1, b1, acc11);
    acc02 = wmma4(a0, b2, acc02);
    acc12 = wmma4(a1, b2, acc12);
    acc03 = wmma4(a0, b3, acc03);
    acc13 = wmma4(a1, b3, acc13);
  }

  v8f acc[2][4] = {{acc00, acc01, acc02, acc03}, {acc10, acc11, acc12, acc13}};
  if (R) {
    #pragma unroll
    for (int mt = 0; mt < 2; mt++)
      #pragma unroll
      for (int c = 0; c < 4; c++)
        #pragma unroll
        for (int r = 0; r < 8; r++) {
          size_t idx = (size_t)(m0 + mt * 16 + r + 8 * hi) * ldc + (n0 + c * 16 + nlo);
          C[idx] = R[idx] + acc[mt][c][r];
        }
  } else {
    #pragma unroll
    for (int mt = 0; mt < 2; mt++)
      #pragma unroll
      for (int c = 0; c < 4; c++)
        #pragma unroll
        for (int r = 0; r < 8; r++) {
          size_t idx = (size_t)(m0 + mt * 16 + r + 8 * hi) * ldc + (n0 + c * 16 + nlo);
          C[idx] = acc[mt][c][r];
        }
  }
}

// ------------------------- sliding-window attention -------------------------
// grid = (T/16, H), block = 32 (one wave). qkv: [T, 3*H*DH], out: [T, H*DH]
__global__ __launch_bounds__(32) void attn_kernel(const float* __restrict__ qkv,
                                                  float* __restrict__ out) {
  const int qb = blockIdx.x * 16;
  const int hd = blockIdx.y;
  const int lane = threadIdx.x;
  const int nlo = lane & 15;
  const int hi = lane >> 4;
  const int koff = hi << 1;
  const int LDQ = 3 * NH * DH;

  const float* Qp = qkv + hd * DH;
  const float* Kp = qkv + NH * DH + hd * DH;
  const float* Vp = qkv + 2 * NH * DH + hd * DH;

  v2f qf[16];
  #pragma unroll
  for (int c = 0; c < 16; c++)
    qf[c] = *(const v2f*)(Qp + (size_t)(qb + nlo) * LDQ + c * 4 + koff);

  v8f oacc[4];
  v8f z = {0,0,0,0,0,0,0,0};
  oacc[0] = z; oacc[1] = z; oacc[2] = z; oacc[3] = z;
  float mrow[8], lrow[8];
  #pragma unroll
  for (int r = 0; r < 8; r++) { mrow[r] = -1e30f; lrow[r] = 0.0f; }

  __shared__ float ps[16][18];

  int j0 = qb - (WIN - 1);
  if (j0 < 0) j0 = 0;
  j0 &= ~15;

  for (int kt = j0; kt <= qb; kt += 16) {
    v8f s = z;
    #pragma unroll
    for (int c = 0; c < 16; c++) {
      v2f kf = *(const v2f*)(Kp + (size_t)(kt + nlo) * LDQ + c * 4 + koff);
      s = wmma4(qf[c], kf, s);
    }
    float pv[8];
    #pragma unroll
    for (int r = 0; r < 8; r++) {
      int i = qb + r + 8 * hi;
      int j = kt + nlo;
      int d = i - j;
      bool ok = (d >= 0) && (d < WIN);
      float sv = ok ? s[r] * 0.125f : -1e30f;   // DH^-0.5 = 1/8
      float tm = sv;
      tm = fmaxf(tm, __shfl_xor(tm, 1, 32));
      tm = fmaxf(tm, __shfl_xor(tm, 2, 32));
      tm = fmaxf(tm, __shfl_xor(tm, 4, 32));
      tm = fmaxf(tm, __shfl_xor(tm, 8, 32));
      float mnew = fmaxf(mrow[r], tm);
      float p = ok ? __expf(sv - mnew) : 0.0f;
      float corr = __expf(mrow[r] - mnew);
      float tsum = p;
      tsum += __shfl_xor(tsum, 1, 32);
      tsum += __shfl_xor(tsum, 2, 32);
      tsum += __shfl_xor(tsum, 4, 32);
      tsum += __shfl_xor(tsum, 8, 32);
      lrow[r] = lrow[r] * corr + tsum;
      mrow[r] = mnew;
      pv[r] = p;
      oacc[0][r] *= corr; oacc[1][r] *= corr; oacc[2][r] *= corr; oacc[3][r] *= corr;
    }
    __syncthreads();
    #pragma unroll
    for (int r = 0; r < 8; r++) ps[r + 8 * hi][nlo] = pv[r];
    __syncthreads();

    v2f af[4];
    #pragma unroll
    for (int kc = 0; kc < 4; kc++)
      af[kc] = *(const v2f*)(&ps[nlo][kc * 4 + koff]);

    #pragma unroll
    for (int kc = 0; kc < 4; kc++) {
      const float* vr = Vp + (size_t)(kt + kc * 4 + koff) * LDQ + nlo;
      #pragma unroll
      for (int c = 0; c < 4; c++) {
        v2f b; b.x = vr[c * 16]; b.y = vr[c * 16 + LDQ];
        oacc[c] = wmma4(af[kc], b, oacc[c]);
      }
    }
  }

  #pragma unroll
  for (int r = 0; r < 8; r++) {
    float inv = 1.0f / lrow[r];
    size_t row = (size_t)(qb + r + 8 * hi) * (NH * DH) + hd * DH + nlo;
    out[row +  0] = oacc[0][r] * inv;
    out[row + 16] = oacc[1][r] * inv;
    out[row + 32] = oacc[2][r] * inv;
    out[row + 48] = oacc[3][r] * inv;
  }
}

// ------------------------- Hebbian memory -------------------------
// reads[j,m] = 1/(8*sqrt(T)) * sum_{i<=j} exp(min(cum[j]-cum[i],0)) * (vals[i].queries[j]) * keys[i,m]
// grid = T/16, block = 32
__global__ __launch_bounds__(32) void hebb_kernel(const float* __restrict__ vals,
                                                  const float* __restrict__ qrs,
                                                  const float* __restrict__ keys,
                                                  const float* __restrict__ cum,
                                                  float* __restrict__ reads) {
  const int jt = blockIdx.x * 16;
  const int lane = threadIdx.x;
  const int nlo = lane & 15;
  const int hi = lane >> 4;
  const int koff = hi << 1;

  v2f qf[16];
  #pragma unroll
  for (int c = 0; c < 16; c++)
    qf[c] = *(const v2f*)(qrs + (size_t)(jt + nlo) * DM + c * 4 + koff);

  const float cj = cum[jt + nlo];

  v8f z = {0,0,0,0,0,0,0,0};
  v8f oacc[4]; oacc[0] = z; oacc[1] = z; oacc[2] = z; oacc[3] = z;

  __shared__ float ps[16][18];

  for (int it = 0; it <= jt; it += 16) {
    v8f s = z;
    #pragma unroll
    for (int c = 0; c < 16; c++) {
      v2f a = *(const v2f*)(vals + (size_t)(it + nlo) * DM + c * 4 + koff);
      s = wmma4(a, qf[c], s);
    }
    __syncthreads();
    #pragma unroll
    for (int r = 0; r < 8; r++) {
      int i = it + r + 8 * hi;
      int j = jt + nlo;
      float ld = fminf(cj - cum[i], 0.0f);
      float p = (j >= i) ? s[r] * __expf(ld) : 0.0f;
      ps[r + 8 * hi][nlo] = p;
    }
    __syncthreads();
    #pragma unroll
    for (int kc = 0; kc < 4; kc++) {
      v2f a; a.x = ps[kc * 4 + koff][nlo]; a.y = ps[kc * 4 + koff + 1][nlo];  // A = P^T
      const float* kr = keys + (size_t)(it + kc * 4 + koff) * DM + nlo;
      #pragma unroll
      for (int c = 0; c < 4; c++) {
        v2f b; b.x = kr[c * 16]; b.y = kr[c * 16 + DM];
        oacc[c] = wmma4(a, b, oacc[c]);
      }
    }
  }

  const float scale = 1.0f / (8.0f * 45.254833995939045f);  // 1/(sqrt(DM)*sqrt(T))
  #pragma unroll
  for (int r = 0; r < 8; r++) {
    size_t row = (size_t)(jt + r + 8 * hi) * DM + nlo;
    reads[row +  0] = oacc[0][r] * scale;
    reads[row + 16] = oacc[1][r] * scale;
    reads[row + 32] = oacc[2][r] * scale;
    reads[row + 48] = oacc[3][r] * scale;
  }
}

// ------------------------- elementwise helpers -------------------------
__global__ __launch_bounds__(256) void silu_mul_kernel(const float* __restrict__ g,
                                                       const float* __restrict__ u,
                                                       float* __restrict__ o, int n) {
  int idx = blockIdx.x * 256 + threadIdx.x;
  if (idx < n) { float x = g[idx]; o[idx] = x * sigf(x) * u[idx]; }
}

__global__ __launch_bounds__(256) void sigmoid_mul_kernel(const float* __restrict__ g, int ldg,
                                                          const float* __restrict__ v, int ldv,
                                                          float* __restrict__ o, int ldo) {
  int idx = blockIdx.x * 256 + threadIdx.x;
  if (idx >= TT * DD) return;
  int t = idx >> 9, d = idx & 511;
  o[(size_t)t * ldo + d] = sigf(g[(size_t)t * ldg + d]) * v[(size_t)t * ldv + d];
}

__global__ __launch_bounds__(256) void dwconv_kernel(const float* __restrict__ in, int ldin,
                                                     const float* __restrict__ w,
                                                     float* __restrict__ out, int ldo) {
  int idx = blockIdx.x * 256 + threadIdx.x;
  if (idx >= TT * DD) return;
  int t = idx >> 9, d = idx & 511;
  float acc = 0.0f;
  #pragma unroll
  for (int k = 0; k < KC; k++) {
    int ts = t - (KC - 1) + k;
    if (ts >= 0) acc += in[(size_t)ts * ldin + d] * w[d * KC + k];
  }
  out[(size_t)t * ldo + d] = acc;
}

__global__ __launch_bounds__(256) void rowgate_kernel(const float* __restrict__ hn,
                                                      const float* __restrict__ gw,
                                                      const float* __restrict__ gb,
                                                      float* __restrict__ gg) {
  const int t = blockIdx.x, i = threadIdx.x;
  const float* row = hn + (size_t)t * DD;
  float s = block_sum_256(row[i] * gw[i] + row[i + 256] * gw[i + 256]);
  if (i == 0) gg[t] = sigf(s + gb[0]);
}

__global__ void cumsum_kernel(const float* __restrict__ gg, float* __restrict__ cum) {
  if (threadIdx.x == 0 && blockIdx.x == 0) {
    float c = 0.0f;
    for (int t = 0; t < TT; t++) { c += logf(gg[t] + 1e-8f); cum[t] = c; }
  }
}

// ------------------------- host launch -------------------------
extern "C" void kernel_launch(void* const* d_in, const int* in_sizes, int n_in,
                              void* d_out, int out_size, void* d_ws, size_t ws_size,
                              hipStream_t stream) {
  (void)in_sizes; (void)n_in; (void)out_size; (void)ws_size;
  auto f = [&](int i) { return (const float*)d_in[i]; };
  const int* x_ids = (const int*)d_in[0];
  const float* embed = f(1);
  const float* ln_in = f(2);
  const float* ln_out = f(3);
  static const int base[6] = {4, 12, 27, 35, 50, 58};

  float* ws = (float*)d_ws;
  float* h     = ws;                       // T*D
  float* hn    = h + (size_t)TT * DD;      // T*D
  float* t1    = hn + (size_t)TT * DD;     // T*2048
  float* t2    = t1 + (size_t)TT * DFF;    // T*2048
  float* t3    = t2 + (size_t)TT * DFF;    // T*2048
  float* keysb = t3 + (size_t)TT * DFF;    // T*64
  float* valsb = keysb + (size_t)TT * DM;  // T*64
  float* qrsb  = valsb + (size_t)TT * DM;  // T*64
  float* readb = qrsb + (size_t)TT * DM;   // T*64
  float* ggb   = readb + (size_t)TT * DM;  // T
  float* cumb  = ggb + TT;                 // T

  auto gemm = [&](const float* A, int lda, const float* W, int ldw,
                  float* C, int ldc, const float* R, int N, int K) {
    dim3 g(TT / 256, N / 64);
    gemm_kernel<<<g, 256, 0, stream>>>(A, lda, W, ldw, C, ldc, R, K);
  };
  auto rms = [&](const float* x, const float* w, float* o) {
    rms_kernel<<<TT, 256, 0, stream>>>(x, w, o);
  };
  auto ffn = [&](const float* Wg, const float* Wu, const float* Wo) {
    gemm(hn, DD, Wg, DD, t1, DFF, nullptr, DFF, DD);
    gemm(hn, DD, Wu, DD, t2, DFF, nullptr, DFF, DD);
    silu_mul_kernel<<<(TT * DFF) / 256, 256, 0, stream>>>(t1, t2, t3, TT * DFF);
    gemm(t3, DFF, Wo, DFF, h, DD, h, DD, DFF);
  };

  // h = rms(embed[x], ln_in)
  embed_rms_kernel<<<TT, 256, 0, stream>>>(x_ids, embed, ln_in, h);

  for (int l = 0; l < 6; l++) {
    int b = base[l];
    const float *Wg = f(b + 0), *Wu = f(b + 1), *Wo = f(b + 2);
    const float *ln1 = f(b + 3), *ln2 = f(b + 4);
    if ((l & 1) == 0) {
      // ---- attention block ----
      const float *qkvw = f(b + 5), *outw = f(b + 6), *gatew = f(b + 7);
      rms(h, ln1, hn);
      gemm(hn, DD, qkvw, DD, t1, 3 * NH * DH, nullptr, 3 * NH * DH, DD);   // qkv
      attn_kernel<<<dim3(TT / 16, NH), 32, 0, stream>>>(t1, t3);            // attn out
      gemm(hn, DD, gatew, DD, t2, DD, nullptr, DD, DD);                     // gate pre
      sigmoid_mul_kernel<<<(TT * DD) / 256, 256, 0, stream>>>(t2, DD, t3, DD, t1, DD);
      gemm(t1, DD, outw, DD, h, DD, h, DD, DD);                             // h += merged @ out^T
      rms(h, ln2, hn);
      ffn(Wg, Wu, Wo);
    } else {
      // ---- hebbian block ----
      const float *mixu = f(b + 5), *mixd = f(b + 6), *convw = f(b + 7);
      const float *lnm = f(b + 8), *keyw = f(b + 9), *valw = f(b + 10);
      const float *qw = f(b + 11), *gw = f(b + 12), *gb = f(b + 13), *memo = f(b + 14);
      rms(h, ln1, hn);
      gemm(hn, DD, mixu, DD, t1, 2 * DD, nullptr, 2 * DD, DD);              // gv = [gate|val]
      dwconv_kernel<<<(TT * DD) / 256, 256, 0, stream>>>(t1 + DD, 2 * DD, convw, t2, DD);
      sigmoid_mul_kernel<<<(TT * DD) / 256, 256, 0, stream>>>(t1, 2 * DD, t2, DD, t3, DD);
      gemm(t3, DD, mixd, DD, h, DD, h, DD, DD);                             // h += (conv*gate)@mixd^T
      rms(h, lnm, hn);
      gemm(hn, DD, keyw, DD, keysb, DM, nullptr, DM, DD);
      gemm(hn, DD, valw, DD, valsb, DM, nullptr, DM, DD);
      gemm(hn, DD, qw, DD, qrsb, DM, nullptr, DM, DD);
      rowgate_kernel<<<TT, 256, 0, stream>>>(hn, gw, gb, ggb);
      cumsum_kernel<<<1, 32, 0, stream>>>(ggb, cumb);
      hebb_kernel<<<TT / 16, 32, 0, stream>>>(valsb, qrsb, keysb, cumb, readb);
      gemm(readb, DM, memo, DM, h, DD, h, DD, DM);                          // h += reads@mem_out^T
      rms(h, ln2, hn);
      ffn(Wg, Wu, Wo);
    }
  }

  // logits = rms(h, ln_out) @ embed^T
  rms(h, ln_out, hn);
  gemm(hn, DD, embed, DD, (float*)d_out, VOCAB, nullptr, VOCAB, DD);
}